// ClusteringLayer_52639119180009
// MI455X (gfx1250) — compile-verified
//
#include <hip/hip_runtime.h>

typedef __attribute__((ext_vector_type(2))) float v2f;
typedef __attribute__((ext_vector_type(4))) float v4f;
typedef __attribute__((ext_vector_type(8))) float v8f;

#define DDIM 512
#define KCLUST 128
#define ROWS_PER_WG 256          // 8 waves x 32 rows
#define CHUNK 32
#define NCHUNK (DDIM / CHUNK)    // 16
#define XS_STRIDE 36             // 32 cols + 4 pad (conflict-free A-frag b64 reads)
#define CS_GSTRIDE 516           // 128*4 + 4 pad (conflict-free B-frag b64 reads)

// v_rcp_f32 + one Newton step: ~0.5 ulp, no div scaffolding. Safe here since
// operands are in [~1e-3, 1e3] (1+d2 >= 1; row sums of positives).
__device__ __forceinline__ float fast_rcp(float x) {
  float r = __builtin_amdgcn_rcpf(x);
  r = r * __builtin_fmaf(-x, r, 2.0f);
  return r;
}

// --- tiny pre-kernel: c2[n] = ||clusters_n||^2 -------------------------------
__global__ void c2_kernel(const float* __restrict__ clusters,
                          float* __restrict__ c2g) {
  const int n = blockIdx.x;
  const int l = threadIdx.x;
  float s = 0.f;
  for (int j = l; j < DDIM; j += 32) {
    const float v = clusters[n * DDIM + j];
    s += v * v;
  }
  #pragma unroll
  for (int mask = 16; mask > 0; mask >>= 1) s += __shfl_xor(s, mask, 32);
  if (l == 0) c2g[n] = s;
}

// --- main kernel: q = normalize_rows( 1 / (1 + ||x-c||^2) ) ------------------
__global__ __launch_bounds__(256) void cluster_q_kernel(
    const float* __restrict__ x, const float* __restrict__ clusters,
    const float* __restrict__ c2g, float* __restrict__ out) {
  __shared__ float xs[ROWS_PER_WG * XS_STRIDE];   // 36.0 KB x tile
  __shared__ float cs[8 * CS_GSTRIDE];            // 16.5 KB clusters tile (frag-major)
  __shared__ float x2s[ROWS_PER_WG];              //  1.0 KB

  const int tid  = threadIdx.x;
  const int lane = tid & 31;
  const int w    = tid >> 5;              // wave id 0..7 (32 rows each)
  const int m    = lane & 15;
  const int off0 = (lane < 16) ? 0 : 2;   // K offset per half-wave (ISA A/B layout)
  const int srow = tid >> 3;              // staging: row-in-pass 0..31
  const int sg   = tid & 7;               // staging: 16B segment / k-group
  const int scol = sg * 4;                // staging: col 0,4,...,28
  const long rowbase = (long)blockIdx.x * ROWS_PER_WG;

  v8f acc[2][8] = {};                     // 2 M-tiles x 8 N-tiles of 16x16 f32
  float accx2[8];
  #pragma unroll
  for (int p = 0; p < 8; ++p) accx2[p] = 0.f;

  for (int c = 0; c < NCHUNK; ++c) {
    const int kk0 = c * CHUNK;
    __syncthreads();                      // previous compute done before restage
    // stage x block (256 rows x 32 cols): one b128 per thread per pass, coalesced
    #pragma unroll
    for (int p = 0; p < 8; ++p) {
      const int r = p * 32 + srow;
      const v4f v = __builtin_nontemporal_load(
          (const v4f*)&x[(rowbase + r) * DDIM + kk0 + scol]);
      *(v4f*)&xs[r * XS_STRIDE + scol] = v;
      accx2[p] += v.x * v.x + v.y * v.y + v.z * v.z + v.w * v.w;
    }
    // stage clusters block (128 x 32) into WMMA-B fragment order: the 4 k's of
    // one lane all belong to k-group sg -> contiguous b128 store
    #pragma unroll
    for (int p = 0; p < 4; ++p) {
      const int n = p * 32 + srow;
      const v4f v = *(const v4f*)&clusters[n * DDIM + kk0 + scol];
      *(v4f*)&cs[sg * CS_GSTRIDE + n * 4] = v;
    }
    __syncthreads();
    // compute: 8 K-steps x 8 N-tiles x 2 M-tiles (B frag reused by both M-tiles)
    const float* xrow0 = &xs[(w * 32 + m) * XS_STRIDE];
    const float* xrow1 = xrow0 + 16 * XS_STRIDE;
    #pragma unroll
    for (int g = 0; g < 8; ++g) {
      const v2f a0 = *(const v2f*)&xrow0[g * 4 + off0];
      const v2f a1 = *(const v2f*)&xrow1[g * 4 + off0];
      const float* csg = &cs[g * CS_GSTRIDE + off0];
      #pragma unroll
      for (int t = 0; t < 8; ++t) {
        const v2f b = *(const v2f*)&csg[(t * 16 + m) * 4];
        acc[0][t] = __builtin_amdgcn_wmma_f32_16x16x4_f32(
            false, a0, false, b, (short)0, acc[0][t], false, false);
        acc[1][t] = __builtin_amdgcn_wmma_f32_16x16x4_f32(
            false, a1, false, b, (short)0, acc[1][t], false, false);
      }
    }
  }

  // finalize ||x||^2: sum over the 8 lanes (tid&7) that shared each row
  #pragma unroll
  for (int p = 0; p < 8; ++p) {
    float s = accx2[p];
    s += __shfl_xor(s, 1, 32);
    s += __shfl_xor(s, 2, 32);
    s += __shfl_xor(s, 4, 32);
    if (sg == 0) x2s[p * 32 + srow] = s;
  }
  __syncthreads();

  // epilogue: d2 = max(x2 + c2 - 2*cross, 0); q = 1/(1+d2); row-normalize
  const int hi = lane >> 4;               // C layout: M = i + 8*hi
  float c2v[8];
  #pragma unroll
  for (int t = 0; t < 8; ++t) c2v[t] = c2g[t * 16 + m];   // N = 16t + m
  #pragma unroll
  for (int mt = 0; mt < 2; ++mt) {
    float x2v[8];
    #pragma unroll
    for (int i = 0; i < 8; ++i) x2v[i] = x2s[w * 32 + mt * 16 + hi * 8 + i];
    #pragma unroll
    for (int t = 0; t < 8; ++t) {
      #pragma unroll
      for (int i = 0; i < 8; ++i) {
        float d2 = fmaxf(x2v[i] + c2v[t] - 2.0f * acc[mt][t][i], 0.0f);
        acc[mt][t][i] = fast_rcp(1.0f + d2);  // alpha=1 -> exponent is 1
      }
    }
    #pragma unroll
    for (int i = 0; i < 8; ++i) {
      float s = acc[mt][0][i];
      #pragma unroll
      for (int t = 1; t < 8; ++t) s += acc[mt][t][i];
      s += __shfl_xor(s, 1, 32);          // reduce within 16-lane half
      s += __shfl_xor(s, 2, 32);
      s += __shfl_xor(s, 4, 32);
      s += __shfl_xor(s, 8, 32);
      const float invs = fast_rcp(s);
      const long gm = rowbase + w * 32 + mt * 16 + hi * 8 + i;
      #pragma unroll
      for (int t = 0; t < 8; ++t) {
        __builtin_nontemporal_store(acc[mt][t][i] * invs,
                                    &out[gm * KCLUST + t * 16 + m]);
      }
    }
  }
}

extern "C" void kernel_launch(void* const* d_in, const int* in_sizes, int n_in,
                              void* d_out, int out_size, void* d_ws,
                              size_t ws_size, hipStream_t stream) {
  (void)n_in; (void)out_size; (void)ws_size;
  const float* x        = (const float*)d_in[0];
  const float* clusters = (const float*)d_in[1];
  float* out = (float*)d_out;
  float* c2g = (float*)d_ws;              // 128 floats of scratch
  const int N = in_sizes[0] / DDIM;       // 131072
  c2_kernel<<<KCLUST, 32, 0, stream>>>(clusters, c2g);
  cluster_q_kernel<<<N / ROWS_PER_WG, 256, 0, stream>>>(x, clusters, c2g, out);
}